// GCN_shared_80582176407949
// MI455X (gfx1250) — compile-verified
//
#include <hip/hip_runtime.h>
#include <hip/hip_bf16.h>
#include <math.h>

#define NN   50000
#define EE   600000
#define FIN  64
#define FHID 128
#define FOUT 128
#define BNEPS 1e-5f

typedef __attribute__((ext_vector_type(2))) float v2f;
typedef __attribute__((ext_vector_type(8))) float v8f;

__device__ __forceinline__ void atomAddF(float* p, float v) {
    unsafeAtomicAdd(p, v);   // lowers to global_atomic_add_f32 on gfx1250
}

// ---------------- utility ----------------
__global__ void zero_f32_kernel(float* __restrict__ p, int n) {
    int i = blockIdx.x * blockDim.x + threadIdx.x;
    if (i < n) p[i] = 0.0f;
}

// count incoming edges per node (SAGE cnt; GCN deg = cnt + 1 self loop)
__global__ void count_edges_kernel(const int* __restrict__ dst, float* __restrict__ cnt) {
    int e = blockIdx.x * blockDim.x + threadIdx.x;
    if (e < EE) atomAddF(&cnt[dst[e]], 1.0f);
}

// dis[i] = rsqrt(cnt+1)  (GCN norm);  cnt[i] <- 1/max(cnt,1)  (SAGE mean)
__global__ void node_norms_kernel(float* __restrict__ cntbuf, float* __restrict__ dis) {
    int i = blockIdx.x * blockDim.x + threadIdx.x;
    if (i < NN) {
        float c = cntbuf[i];
        dis[i] = rsqrtf(c + 1.0f);
        cntbuf[i] = 1.0f / fmaxf(c, 1.0f);
    }
}

// ---------------- GEMM 1: h1 = x @ W1  (exact fp32 WMMA) ----------------
// grid = 3125 (M tiles), block = 256 (8 waves -> 8 col tiles of 16)
__global__ __launch_bounds__(256) void gemm1_wmma_kernel(
        const float* __restrict__ X, const float* __restrict__ W,
        float* __restrict__ H) {
    int wave = threadIdx.x >> 5;
    int lane = threadIdx.x & 31;
    int half = lane >> 4;          // 0: K pair {0,1}, 1: K pair {2,3}
    int l    = lane & 15;
    int row0 = blockIdx.x * 16;
    int col0 = wave * 16;
    const float* xrow = X + (size_t)(row0 + l) * FIN;
    v8f acc = {};
    for (int k = 0; k < FIN; k += 4) {
        int kk = k + 2 * half;
        v2f a, b;
        a[0] = xrow[kk + 0];
        a[1] = xrow[kk + 1];
        b[0] = W[(size_t)(kk + 0) * FHID + col0 + l];
        b[1] = W[(size_t)(kk + 1) * FHID + col0 + l];
        acc = __builtin_amdgcn_wmma_f32_16x16x4_f32(false, a, false, b,
                                                    (short)0, acc, false, false);
    }
    int roff = half * 8;
    for (int r = 0; r < 8; ++r)
        H[(size_t)(row0 + r + roff) * FHID + col0 + l] = acc[r];
}

// ---------------- GCN aggregation ----------------
// self-loop contribution + bias: agg[i] = h1[i]*dis[i]^2 + b1
__global__ void gcn_self_init_kernel(const float* __restrict__ h1,
                                     const float* __restrict__ dis,
                                     const float* __restrict__ b1,
                                     float* __restrict__ agg) {
    int idx = blockIdx.x * blockDim.x + threadIdx.x;   // over N * 32 float4 chunks
    if (idx >= NN * 32) return;
    int i  = idx >> 5;
    int f4 = idx & 31;
    float d = dis[i];
    float w = d * d;
    float4 v = ((const float4*)(h1 + (size_t)i * FHID))[f4];
    float4 bb = ((const float4*)b1)[f4];
    float4 o;
    o.x = v.x * w + bb.x; o.y = v.y * w + bb.y;
    o.z = v.z * w + bb.z; o.w = v.w * w + bb.w;
    ((float4*)(agg + (size_t)i * FHID))[f4] = o;
}

// one wave per edge: agg[dst] += h1[src] * dis[src]*dis[dst]
__global__ __launch_bounds__(256) void gcn_edge_scatter_kernel(
        const int* __restrict__ src, const int* __restrict__ dst,
        const float* __restrict__ h1, const float* __restrict__ dis,
        float* __restrict__ agg) {
    int e    = (blockIdx.x * blockDim.x + threadIdx.x) >> 5;
    int lane = threadIdx.x & 31;
    if (e >= EE) return;
    int s = src[e], d = dst[e];
    float w = dis[s] * dis[d];
    float4 v = ((const float4*)(h1 + (size_t)s * FHID))[lane];
    float* out = agg + (size_t)d * FHID + lane * 4;
    atomAddF(out + 0, v.x * w);
    atomAddF(out + 1, v.y * w);
    atomAddF(out + 2, v.z * w);
    atomAddF(out + 3, v.w * w);
}

// ---------------- SAGE neighbor sum: neigh[dst] += h[src] ----------------
__global__ __launch_bounds__(256) void sage_edge_scatter_kernel(
        const int* __restrict__ src, const int* __restrict__ dst,
        const float* __restrict__ h, float* __restrict__ neigh) {
    int e    = (blockIdx.x * blockDim.x + threadIdx.x) >> 5;
    int lane = threadIdx.x & 31;
    if (e >= EE) return;
    int s = src[e], d = dst[e];
    float4 v = ((const float4*)(h + (size_t)s * FHID))[lane];
    float* out = neigh + (size_t)d * FHID + lane * 4;
    atomAddF(out + 0, v.x);
    atomAddF(out + 1, v.y);
    atomAddF(out + 2, v.z);
    atomAddF(out + 3, v.w);
}

// ---------------- GEMM 2 (fused): out = (neigh*invc) @ Wl + h @ Wr + bl ----------------
__global__ __launch_bounds__(256) void gemm2_wmma_kernel(
        const float* __restrict__ NEI, const float* __restrict__ INVC,
        const float* __restrict__ H,   const float* __restrict__ Wl,
        const float* __restrict__ Wr,  const float* __restrict__ BL,
        float* __restrict__ OUT) {
    int wave = threadIdx.x >> 5;
    int lane = threadIdx.x & 31;
    int half = lane >> 4;
    int l    = lane & 15;
    int row0 = blockIdx.x * 16;
    int col0 = wave * 16;
    int rowA = row0 + l;
    float inv = INVC[rowA];
    const float* nrow = NEI + (size_t)rowA * FHID;
    const float* hrow = H   + (size_t)rowA * FHID;
    v8f acc = {};
    for (int k = 0; k < FHID; k += 4) {
        int kk = k + 2 * half;
        v2f a, b;
        // mean-aggregated neighbors through lin_l
        a[0] = nrow[kk + 0] * inv;
        a[1] = nrow[kk + 1] * inv;
        b[0] = Wl[(size_t)(kk + 0) * FOUT + col0 + l];
        b[1] = Wl[(size_t)(kk + 1) * FOUT + col0 + l];
        acc = __builtin_amdgcn_wmma_f32_16x16x4_f32(false, a, false, b,
                                                    (short)0, acc, false, false);
        // root features through lin_r
        a[0] = hrow[kk + 0];
        a[1] = hrow[kk + 1];
        b[0] = Wr[(size_t)(kk + 0) * FOUT + col0 + l];
        b[1] = Wr[(size_t)(kk + 1) * FOUT + col0 + l];
        acc = __builtin_amdgcn_wmma_f32_16x16x4_f32(false, a, false, b,
                                                    (short)0, acc, false, false);
    }
    float bias = BL[col0 + l];
    int roff = half * 8;
    for (int r = 0; r < 8; ++r)
        OUT[(size_t)(row0 + r + roff) * FOUT + col0 + l] = acc[r] + bias;
}

// ---------------- BatchNorm (training-mode batch stats) ----------------
// 200 blocks x 128 threads; block b reduces rows [b*250, b*250+250), thread = column
__global__ __launch_bounds__(128) void bn_stats_kernel(
        const float* __restrict__ in, float* __restrict__ sums,
        float* __restrict__ sumsq) {
    int col = threadIdx.x;
    int r0  = blockIdx.x * 250;
    float s = 0.0f, q = 0.0f;
    for (int r = 0; r < 250; ++r) {
        float v = in[(size_t)(r0 + r) * FHID + col];
        s += v;
        q += v * v;
    }
    atomAddF(&sums[col], s);
    atomAddF(&sumsq[col], q);
}

// sums <- mean, sumsq <- rsqrt(var + eps)
__global__ void bn_finalize_kernel(float* __restrict__ sums, float* __restrict__ sumsq) {
    int c = threadIdx.x;
    float m   = sums[c] * (1.0f / NN);
    float var = sumsq[c] * (1.0f / NN) - m * m;
    sums[c]  = m;
    sumsq[c] = rsqrtf(fmaxf(var, 0.0f) + BNEPS);
}

__global__ void bn_apply_relu_kernel(const float* __restrict__ in, float* __restrict__ out,
                                     const float* __restrict__ mean,
                                     const float* __restrict__ invstd,
                                     const float* __restrict__ gamma,
                                     const float* __restrict__ beta) {
    int idx = blockIdx.x * blockDim.x + threadIdx.x;
    if (idx >= NN * FHID) return;
    int c = idx & (FHID - 1);
    float v = in[idx];
    float y = gamma[c] * (v - mean[c]) * invstd[c] + beta[c];
    out[idx] = fmaxf(y, 0.0f);
}

// ---------------- host-side orchestration ----------------
extern "C" void kernel_launch(void* const* d_in, const int* in_sizes, int n_in,
                              void* d_out, int out_size, void* d_ws, size_t ws_size,
                              hipStream_t stream) {
    const float* x     = (const float*)d_in[0];
    const int*   ei    = (const int*)d_in[1];
    const float* W1    = (const float*)d_in[2];
    const float* b1    = (const float*)d_in[3];
    const float* g1    = (const float*)d_in[4];
    const float* be1   = (const float*)d_in[5];
    const float* Wl    = (const float*)d_in[6];
    const float* bl    = (const float*)d_in[7];
    const float* Wr    = (const float*)d_in[8];
    const float* g2    = (const float*)d_in[9];
    const float* be2   = (const float*)d_in[10];
    const int* srcp = ei;
    const int* dstp = ei + EE;
    float* out = (float*)d_out;

    // workspace layout (floats): ~51.6 MB
    float* ws    = (float*)d_ws;
    float* h1    = ws;                         // [N,128] GEMM1 out; later reused as neigh
    float* agg   = ws + (size_t)NN * FHID;     // [N,128] GCN agg; becomes h after BN1
    float* cnt   = ws + 2 * (size_t)NN * FHID; // [N] edge count -> 1/max(cnt,1)
    float* dis   = cnt + NN;                   // [N] rsqrt(deg)
    float* sums  = dis + NN;                   // [128]
    float* sumsq = sums + 128;                 // [128]
    float* neigh = h1;                         // alias: h1 dead after GCN scatter

    const int T = 256;
    // degree / count
    zero_f32_kernel<<<(NN + T - 1) / T, T, 0, stream>>>(cnt, NN);
    count_edges_kernel<<<(EE + T - 1) / T, T, 0, stream>>>(dstp, cnt);
    node_norms_kernel<<<(NN + T - 1) / T, T, 0, stream>>>(cnt, dis);

    // layer 1: GCNConv
    gemm1_wmma_kernel<<<NN / 16, T, 0, stream>>>(x, W1, h1);
    gcn_self_init_kernel<<<(NN * 32 + T - 1) / T, T, 0, stream>>>(h1, dis, b1, agg);
    gcn_edge_scatter_kernel<<<(EE * 32 + T - 1) / T, T, 0, stream>>>(srcp, dstp, h1, dis, agg);

    // BN1 + ReLU (in place on agg)
    zero_f32_kernel<<<1, T, 0, stream>>>(sums, 256);
    bn_stats_kernel<<<NN / 250, 128, 0, stream>>>(agg, sums, sumsq);
    bn_finalize_kernel<<<1, 128, 0, stream>>>(sums, sumsq);
    bn_apply_relu_kernel<<<(NN * FHID + T - 1) / T, T, 0, stream>>>(agg, agg, sums, sumsq, g1, be1);

    // layer 2: SAGEConv (mean aggregation)
    zero_f32_kernel<<<((NN * FHID) + T - 1) / T, T, 0, stream>>>(neigh, NN * FHID);
    sage_edge_scatter_kernel<<<(EE * 32 + T - 1) / T, T, 0, stream>>>(srcp, dstp, agg, neigh);
    gemm2_wmma_kernel<<<NN / 16, T, 0, stream>>>(neigh, cnt, agg, Wl, Wr, bl, out);

    // BN2 + ReLU (in place on out)
    zero_f32_kernel<<<1, T, 0, stream>>>(sums, 256);
    bn_stats_kernel<<<NN / 250, 128, 0, stream>>>(out, sums, sumsq);
    bn_finalize_kernel<<<1, 128, 0, stream>>>(sums, sumsq);
    bn_apply_relu_kernel<<<(NN * FHID + T - 1) / T, T, 0, stream>>>(out, out, sums, sumsq, g2, be2);
}